// SinkhornLoss_59906203844618
// MI455X (gfx1250) — compile-verified
//
#include <hip/hip_runtime.h>
#include <math.h>

typedef float v2f __attribute__((ext_vector_type(2)));
typedef float v8f __attribute__((ext_vector_type(8)));

#define NPTS 4096
#define NB   4
#define DIM  64

#define LOG2E 1.4426950408889634f
#define LN2   0.6931471805599453f

__device__ __forceinline__ float fast_exp2(float x) { return __builtin_amdgcn_exp2f(x); }
__device__ __forceinline__ float fast_log2(float x) { return __builtin_amdgcn_logf(x); }

// ---------------------------------------------------------------------------
// Transpose [C][N] -> [N][C] and compute 0.5*|p|^2 per point.
// blockIdx.y: 0 = x, 1 = y ; blockIdx.z = batch
// ---------------------------------------------------------------------------
__global__ __launch_bounds__(256) void sink_transpose_kernel(
    const float* __restrict__ x, const float* __restrict__ y,
    float* __restrict__ xt, float* __restrict__ yt,
    float* __restrict__ nx, float* __restrict__ ny)
{
  const int n = blockIdx.x * blockDim.x + threadIdx.x;
  const int b = blockIdx.z;
  const float* src = blockIdx.y ? y : x;
  float* dstT = blockIdx.y ? yt : xt;
  float* dstN = blockIdx.y ? ny : nx;
  src  += (size_t)b * DIM * NPTS;
  dstT += (size_t)b * NPTS * DIM;
  dstN += (size_t)b * NPTS;
  float acc = 0.f;
#pragma unroll
  for (int c = 0; c < DIM; ++c) {
    float v = src[(size_t)c * NPTS + n];
    dstT[(size_t)n * DIM + c] = v;
    acc = fmaf(v, v, acc);
  }
  dstN[n] = 0.5f * acc;
}

// ---------------------------------------------------------------------------
// Build base-2 scaled h2[j] = LOG2E * ( logw + (pot_scale*pot_src[j]
//                                       - 0.5|p_j|^2) * inv_eps )
// vec: 0=ft(uses g_ab,ny) 1=gt(uses f_ba,nx) 2=fa(f_aa,nx) 3=gb(g_bb,ny)
// inv_eps2 = LOG2E / eps is passed in; logw2 = LOG2E * -log(4096).
// ---------------------------------------------------------------------------
__global__ __launch_bounds__(256) void sink_prep_kernel(
    const float* __restrict__ pot,   // [4][NB][NPTS] current potentials
    const float* __restrict__ nx, const float* __restrict__ ny,
    float* __restrict__ hout,        // [4][NB][NPTS]
    float inv_eps2, float pot_scale)
{
  const int j   = blockIdx.x * blockDim.x + threadIdx.x;
  const int vec = blockIdx.y, b = blockIdx.z;
  const int src = (vec == 0) ? 1 : (vec == 1) ? 0 : vec;
  const float* nrm = (vec == 1 || vec == 2) ? nx : ny;
  const float logw2 = -8.317766166719343f * LOG2E;  // -log(4096) in base 2
  float p = pot_scale * pot[((size_t)src * NB + b) * NPTS + j];
  float h = logw2 + (p - nrm[(size_t)b * NPTS + j]) * inv_eps2;
  hout[((size_t)vec * NB + b) * NPTS + j] = h;
}

// ---------------------------------------------------------------------------
// Fused softmin (base-2 internal): out[i] = alpha*old[i] + beta*( 0.5|r_i|^2
//     - eps * LN2 * LSE2_j( h2[j] + (r_i . c_j) * inv_eps2 ) )
// One wave per 16 rows; WMMA f32 16x16x4; 8 column sub-tiles (128 columns)
// per sweep iteration; online base-2 LSE with v_exp_f32 / v_log_f32.
// ---------------------------------------------------------------------------
__global__ __launch_bounds__(256) void sink_softmin_kernel(
    const float* __restrict__ xt, const float* __restrict__ yt,
    const float* __restrict__ nx, const float* __restrict__ ny,
    const float* __restrict__ hbuf,    // [4][NB][NPTS] base-2 h'
    const float* __restrict__ potOld,  // [4][NB][NPTS]
    float* __restrict__ potNew,        // [4][NB][NPTS]
    float eps, float inv_eps2, float alpha, float beta)
{
  const int vec = blockIdx.y, b = blockIdx.z;
  const float *rowT, *colT, *rn;
  if (vec == 0)      { rowT = xt; colT = yt; rn = nx; }
  else if (vec == 1) { rowT = yt; colT = xt; rn = ny; }
  else if (vec == 2) { rowT = xt; colT = xt; rn = nx; }
  else               { rowT = yt; colT = yt; rn = ny; }
  rowT += (size_t)b * NPTS * DIM;
  colT += (size_t)b * NPTS * DIM;
  rn   += (size_t)b * NPTS;
  const float* hvec = hbuf   + ((size_t)vec * NB + b) * NPTS;
  const float* oldp = potOld + ((size_t)vec * NB + b) * NPTS;
  float*       outp = potNew + ((size_t)vec * NB + b) * NPTS;

  const int tid = threadIdx.x;
  const int wave = tid >> 5, lane = tid & 31;
  const int lane16 = lane & 15, hi = lane >> 4;
  const int i0 = blockIdx.x * 128 + wave * 16;

  // Preload A fragments: rows i0..i0+15, all 64 channels.
  // f32 16x4 A layout: lanes 0-15 hold K={0,1} (v0,v1), lanes 16-31 K={2,3}.
  v2f a[16];
  const float* arow = rowT + (size_t)(i0 + lane16) * DIM + 2 * hi;
#pragma unroll
  for (int k4 = 0; k4 < 16; ++k4)
    a[k4] = *(const v2f*)(arow + 4 * k4);

  float m[8], s[8];
#pragma unroll
  for (int v = 0; v < 8; ++v) { m[v] = -1e30f; s[v] = 0.f; }

  for (int j0 = 0; j0 < NPTS; j0 += 128) {
    v8f acc[8];
#pragma unroll
    for (int t = 0; t < 8; ++t) acc[t] = (v8f){};
    const float* bbase = colT + (size_t)(j0 + lane16) * DIM + 2 * hi;
#pragma unroll
    for (int k4 = 0; k4 < 16; ++k4) {
#pragma unroll
      for (int t = 0; t < 8; ++t) {
        v2f bt = *(const v2f*)(bbase + (size_t)t * 16 * DIM + 4 * k4);
        acc[t] = __builtin_amdgcn_wmma_f32_16x16x4_f32(
            false, a[k4], false, bt, (short)0, acc[t], false, false);
      }
    }
    float h[8];
#pragma unroll
    for (int t = 0; t < 8; ++t) h[t] = hvec[j0 + 16 * t + lane16];
#pragma unroll
    for (int v = 0; v < 8; ++v) {
      float sv[8];
#pragma unroll
      for (int t = 0; t < 8; ++t) sv[t] = fmaf(acc[t][v], inv_eps2, h[t]);
      float t01 = fmaxf(sv[0], sv[1]), t23 = fmaxf(sv[2], sv[3]);
      float t45 = fmaxf(sv[4], sv[5]), t67 = fmaxf(sv[6], sv[7]);
      float tmax = fmaxf(fmaxf(t01, t23), fmaxf(t45, t67));
      float mn = fmaxf(m[v], tmax);
      float e0 = fast_exp2(sv[0] - mn) + fast_exp2(sv[1] - mn);
      float e1 = fast_exp2(sv[2] - mn) + fast_exp2(sv[3] - mn);
      float e2 = fast_exp2(sv[4] - mn) + fast_exp2(sv[5] - mn);
      float e3 = fast_exp2(sv[6] - mn) + fast_exp2(sv[7] - mn);
      float resc = fast_exp2(m[v] - mn);
      s[v] = fmaf(s[v], resc, (e0 + e1) + (e2 + e3));
      m[v] = mn;
    }
  }

  // Cross-lane LSE combine within each 16-lane half (xor masks stay in-half).
#pragma unroll
  for (int v = 0; v < 8; ++v) {
    float mv = m[v], svv = s[v];
#pragma unroll
    for (int off = 1; off < 16; off <<= 1) {
      float om = __shfl_xor(mv, off, 32);
      float os = __shfl_xor(svv, off, 32);
      float mn = fmaxf(mv, om);
      svv = svv * fast_exp2(mv - mn) + os * fast_exp2(om - mn);
      mv = mn;
    }
    m[v] = mv; s[v] = svv;
  }

  // C/D layout: VGPR v <-> row (v + 8*hi). Lanes 0 and 16 write results.
  if (lane16 == 0) {
#pragma unroll
    for (int v = 0; v < 8; ++v) {
      int r = i0 + v + 8 * hi;
      float lse_nat = LN2 * (m[v] + fast_log2(s[v]));
      float val = rn[r] - eps * lse_nat;
      float o = (alpha != 0.f) ? oldp[r] : 0.f;
      outp[r] = alpha * o + beta * val;
    }
  }
}

// ---------------------------------------------------------------------------
// loss = sum_b sum_i (f_ba - f_aa + g_ab - g_bb) / (NPTS*NB)
// ---------------------------------------------------------------------------
__global__ __launch_bounds__(256) void sink_reduce_kernel(
    const float* __restrict__ potF, float* __restrict__ out)
{
  __shared__ float sm[256];
  const int tid = threadIdx.x;
  float acc = 0.f;
  for (int idx = tid; idx < NB * NPTS; idx += 256) {
    size_t o = (size_t)idx;  // [b][i] flattened == b*NPTS+i
    acc += potF[(size_t)0 * NB * NPTS + o] - potF[(size_t)2 * NB * NPTS + o]
         + potF[(size_t)1 * NB * NPTS + o] - potF[(size_t)3 * NB * NPTS + o];
  }
  sm[tid] = acc;
  __syncthreads();
  for (int st = 128; st > 0; st >>= 1) {
    if (tid < st) sm[tid] += sm[tid + st];
    __syncthreads();
  }
  if (tid == 0) out[0] = sm[0] / (float)((size_t)NPTS * NB);
}

// ---------------------------------------------------------------------------
extern "C" void kernel_launch(void* const* d_in, const int* in_sizes, int n_in,
                              void* d_out, int out_size, void* d_ws, size_t ws_size,
                              hipStream_t stream)
{
  (void)in_sizes; (void)n_in; (void)out_size; (void)ws_size;
  const float* x = (const float*)d_in[0];
  const float* y = (const float*)d_in[1];
  float* out = (float*)d_out;

  char* ws = (char*)d_ws;
  size_t off = 0;
  float* xt = (float*)(ws + off); off += (size_t)NB * NPTS * DIM * sizeof(float);
  float* yt = (float*)(ws + off); off += (size_t)NB * NPTS * DIM * sizeof(float);
  float* nx = (float*)(ws + off); off += (size_t)NB * NPTS * sizeof(float);
  float* ny = (float*)(ws + off); off += (size_t)NB * NPTS * sizeof(float);
  float* pot0 = (float*)(ws + off); off += (size_t)4 * NB * NPTS * sizeof(float);
  float* pot1 = (float*)(ws + off); off += (size_t)4 * NB * NPTS * sizeof(float);
  float* hbuf = (float*)(ws + off); off += (size_t)4 * NB * NPTS * sizeof(float);
  float* pot[2] = {pot0, pot1};

  // Stage 1: transpose + norms
  sink_transpose_kernel<<<dim3(NPTS / 256, 2, NB), 256, 0, stream>>>(
      x, y, xt, yt, nx, ny);

  // eps schedule (np.arange(P*log(D), P*log(blur), P*log(scaling)) + [blur^P])
  float eps_list[80]; int n_eps = 0;
  {
    double start = 2.0 * log(16.0), stop = 2.0 * log(0.05), st = 2.0 * log(0.9);
    for (double e = start; e > stop && n_eps < 79; e += st)
      eps_list[n_eps++] = (float)exp(e);
    eps_list[n_eps++] = 0.05f * 0.05f;
  }

  const dim3 pgrid(NPTS / 256, 4, NB);
  const dim3 mgrid(NPTS / 128, 4, NB);
  int cur = 0;

  // Init: softmin at eps0 with zero potentials (pot_scale = 0), write pot[0].
  {
    float e0 = eps_list[0], ie2 = LOG2E / e0;
    sink_prep_kernel<<<pgrid, 256, 0, stream>>>(pot[cur], nx, ny, hbuf, ie2, 0.0f);
    sink_softmin_kernel<<<mgrid, 256, 0, stream>>>(
        xt, yt, nx, ny, hbuf, pot[cur], pot[cur], e0, ie2, 0.0f, 1.0f);
  }

  // Annealing scan over the full eps list with 0.5-averaging.
  for (int t = 0; t < n_eps; ++t) {
    float e = eps_list[t], ie2 = LOG2E / e;
    sink_prep_kernel<<<pgrid, 256, 0, stream>>>(pot[cur], nx, ny, hbuf, ie2, 1.0f);
    sink_softmin_kernel<<<mgrid, 256, 0, stream>>>(
        xt, yt, nx, ny, hbuf, pot[cur], pot[cur ^ 1], e, ie2, 0.5f, 0.5f);
    cur ^= 1;
  }

  // Final extrapolation at eps = blur^P (no averaging).
  {
    float e = eps_list[n_eps - 1], ie2 = LOG2E / e;
    sink_prep_kernel<<<pgrid, 256, 0, stream>>>(pot[cur], nx, ny, hbuf, ie2, 1.0f);
    sink_softmin_kernel<<<mgrid, 256, 0, stream>>>(
        xt, yt, nx, ny, hbuf, pot[cur], pot[cur ^ 1], e, ie2, 0.0f, 1.0f);
    cur ^= 1;
  }

  sink_reduce_kernel<<<1, 256, 0, stream>>>(pot[cur], out);
}